// QNNClassifier_71339406787262
// MI455X (gfx1250) — compile-verified
//
#include <hip/hip_runtime.h>
#include <hip/hip_bf16.h>

#define NQ    14
#define DIM   16384
#define DEPTH 8
#define TPB   256

typedef __attribute__((ext_vector_type(2))) float v2f;
typedef __attribute__((ext_vector_type(8))) float v8f;

// Inverse of the composed CX-ring permutation P = p0∘p1∘...∘p13,
// where p_c: if bit(13-c) set, flip bit(12-c) (c<13), p13: if bit0 set flip bit13.
// Each p_c is an involution, so Pinv applies p0 first ... p13 last.
__device__ __forceinline__ int pinv14(int j) {
#pragma unroll
  for (int c = 0; c < NQ; ++c) {
    const int sc = NQ - 1 - c;
    const int mt = (c < NQ - 1) ? (1 << (NQ - 2 - c)) : (1 << (NQ - 1));
    j ^= (-((j >> sc) & 1)) & mt;
  }
  return j;
}

// Build fused 16x16 complex gate matrices per (layer, group).
// Group 0..2: qubits 4g..4g+3. Group 3: I (x) I (x) M12 (x) M13 on bits 3..0.
// M_q = RY(b) * RZ(a):
//   [[cy*e^{-ia/2}, -sy*e^{+ia/2}], [sy*e^{-ia/2}, cy*e^{+ia/2}]]
__global__ __launch_bounds__(TPB) void qnn_build_gates(const float* __restrict__ params,
                                                       float* __restrict__ Ure,
                                                       float* __restrict__ Uim) {
  const int layer = blockIdx.x >> 2;
  const int group = blockIdx.x & 3;
  const int tid = threadIdx.x;
  const int r = tid >> 4, col = tid & 15;
  float pr = 1.f, pi = 0.f;
#pragma unroll
  for (int k = 0; k < 4; ++k) {
    const int rb = (r >> (3 - k)) & 1;
    const int cb = (col >> (3 - k)) & 1;
    const int q = (group < 3) ? (group * 4 + k) : ((k < 2) ? -1 : (10 + k));
    float mr, mi;
    if (q < 0) {  // identity factor
      mr = (rb == cb) ? 1.f : 0.f;
      mi = 0.f;
    } else {
      const float a = params[layer * 2 * NQ + 2 * q];      // RZ angle
      const float b = params[layer * 2 * NQ + 2 * q + 1];  // RY angle
      const float cy = cosf(b * 0.5f), sy = sinf(b * 0.5f);
      const float cz = cosf(a * 0.5f), sz = sinf(a * 0.5f);
      if (rb == 0) {
        if (cb == 0) { mr =  cy * cz; mi = -cy * sz; }
        else         { mr = -sy * cz; mi = -sy * sz; }
      } else {
        if (cb == 0) { mr =  sy * cz; mi = -sy * sz; }
        else         { mr =  cy * cz; mi =  cy * sz; }
      }
    }
    const float nr = pr * mr - pi * mi;
    const float ni = pr * mi + pi * mr;
    pr = nr; pi = ni;
  }
  const int off = (layer * 4 + group) * 256 + r * 16 + col;
  Ure[off] = pr;
  Uim[off] = pi;
}

// One workgroup per batch element. State (re/im, double-buffered) lives in LDS.
__global__ __launch_bounds__(TPB) void qnn_sim(const float* __restrict__ x,
                                               const float* __restrict__ Ure,
                                               const float* __restrict__ Uim,
                                               float* __restrict__ out) {
  __shared__ float s_re[2][DIM];
  __shared__ float s_im[2][DIM];
  __shared__ float enc_c[16];
  __shared__ float enc_s[16];

  const int tid = threadIdx.x;
  const int lane = tid & 31;
  const int wave = tid >> 5;
  const int b = blockIdx.x;

  // --- feature encoding: product state from RY(x*pi) applied to |0..0> ---
  if (tid < NQ) {
    const float ang = x[b * NQ + tid] * 1.57079632679489662f;  // x*pi/2
    enc_c[tid] = cosf(ang);
    enc_s[tid] = sinf(ang);
  }
  __syncthreads();

  float cq[NQ], sq[NQ];
#pragma unroll
  for (int q = 0; q < NQ; ++q) { cq[q] = enc_c[q]; sq[q] = enc_s[q]; }

  for (int n = tid; n < DIM; n += TPB) {
    float p = 1.f;
#pragma unroll
    for (int q = 0; q < NQ; ++q)
      p *= ((n >> (NQ - 1 - q)) & 1) ? sq[q] : cq[q];
    s_re[0][n] = p;
    s_im[0][n] = 0.f;
  }
  __syncthreads();

  const int jj = lane & 15;  // tile column (N index) / A-row index
  const int hi = lane >> 4;  // lane half selects K pair / M offset
  int cur = 0;

  for (int layer = 0; layer < DEPTH; ++layer) {
#pragma unroll
    for (int g = 0; g < 4; ++g) {
      const int p = (g < 3) ? (10 - 4 * g) : 0;  // low bit position of group
      const int pm = (1 << p) - 1;
      const float* __restrict__ ur = Ure + (layer * 4 + g) * 256;
      const float* __restrict__ ui = Uim + (layer * 4 + g) * 256;

      // A operands (gate matrix), reused for all tiles of this stage.
      // A 16x4 layout: lanes 0-15: (K=k0,k0+1), lanes 16-31: (K=k0+2,k0+3), row M = lane&15.
      v2f aR[4], aI[4], aIn[4];
#pragma unroll
      for (int c0 = 0; c0 < 4; ++c0) {
        const int kk = 4 * c0 + 2 * hi;
        const int ao = jj * 16 + kk;
        aR[c0][0] = ur[ao];     aR[c0][1] = ur[ao + 1];
        aI[c0][0] = ui[ao];     aI[c0][1] = ui[ao + 1];
        aIn[c0][0] = -aI[c0][0]; aIn[c0][1] = -aI[c0][1];
      }

      float* __restrict__ wre = s_re[cur];
      float* __restrict__ wim = s_im[cur];
      float* __restrict__ dre = s_re[cur ^ 1];
      float* __restrict__ dim_ = s_im[cur ^ 1];

      // 64 disjoint 16x16 tiles, 8 per wave. Constant trip count -> scalar
      // loop control, no exec masking: EXEC stays all-ones around every WMMA.
#pragma unroll 1
      for (int it = 0; it < 8; ++it) {
        const int t = it * 8 + wave;
        const int colIdx = t * 16 + jj;
        const int base = ((colIdx >> p) << (p + 4)) | (colIdx & pm);
        v8f accR = {};
        v8f accI = {};
#pragma unroll
        for (int c0 = 0; c0 < 4; ++c0) {
          const int kb = 4 * c0 + 2 * hi;
          const int e0 = base + (kb << p);
          const int e1 = base + ((kb + 1) << p);
          v2f bR, bI;
          bR[0] = wre[e0]; bR[1] = wre[e1];
          bI[0] = wim[e0]; bI[1] = wim[e1];
          // Yr = Ur*Xr - Ui*Xi ; Yi = Ur*Xi + Ui*Xr   (chained f32 WMMA)
          accR = __builtin_amdgcn_wmma_f32_16x16x4_f32(false, aR[c0],  false, bR, (short)0, accR, false, false);
          accR = __builtin_amdgcn_wmma_f32_16x16x4_f32(false, aIn[c0], false, bI, (short)0, accR, false, false);
          accI = __builtin_amdgcn_wmma_f32_16x16x4_f32(false, aR[c0],  false, bI, (short)0, accI, false, false);
          accI = __builtin_amdgcn_wmma_f32_16x16x4_f32(false, aI[c0],  false, bR, (short)0, accI, false, false);
        }
        // D layout: VGPR v -> row M = v + 8*hi, col N = lane&15
        if (g < 3) {
#pragma unroll
          for (int v = 0; v < 8; ++v) {
            const int m = v + 8 * hi;
            const int e = base + (m << p);
            wre[e] = accR[v];
            wim[e] = accI[v];
          }
        } else {
          // Last stage: fold the CX-ring permutation into the scatter:
          // new[Pinv(e)] = (G*s)[e]   (p == 0 here, so e = base + m)
#pragma unroll
          for (int v = 0; v < 8; ++v) {
            const int m = v + 8 * hi;
            const int d = pinv14(base + m);
            dre[d] = accR[v];
            dim_[d] = accI[v];
          }
        }
      }
      __syncthreads();
    }
    cur ^= 1;
  }

  // --- measurement: <Z0> = sum |amp|^2 * (+1 if n<DIM/2 else -1) ---
  float local = 0.f;
  const float* __restrict__ fre = s_re[cur];
  const float* __restrict__ fim = s_im[cur];
  for (int n = tid; n < DIM; n += TPB) {
    const float re = fre[n], im = fim[n];
    const float pr2 = re * re + im * im;
    local += (n < (DIM / 2)) ? pr2 : -pr2;
  }
  float* red = s_re[cur ^ 1];  // free buffer reused as reduction scratch
  red[tid] = local;
  __syncthreads();
  for (int sft = TPB / 2; sft > 0; sft >>= 1) {
    if (tid < sft) red[tid] += red[tid + sft];
    __syncthreads();
  }
  if (tid == 0) out[b] = (red[0] + 1.f) * 0.5f;
}

extern "C" void kernel_launch(void* const* d_in, const int* in_sizes, int n_in,
                              void* d_out, int out_size, void* d_ws, size_t ws_size,
                              hipStream_t stream) {
  const float* x = (const float*)d_in[0];       // [B, 14] f32
  const float* params = (const float*)d_in[1];  // [2*14*8] f32
  float* out = (float*)d_out;                   // [B] f32
  const int B = in_sizes[0] / NQ;

  float* Ure = (float*)d_ws;                    // [8][4][16][16] f32
  float* Uim = Ure + DEPTH * 4 * 256;           // [8][4][16][16] f32

  qnn_build_gates<<<DEPTH * 4, TPB, 0, stream>>>(params, Ure, Uim);
  qnn_sim<<<B, TPB, 0, stream>>>(x, Ure, Uim, out);
}